// ComplexUpSampling2D_13778255086287
// MI455X (gfx1250) — compile-verified
//
#include <hip/hip_runtime.h>
#include <hip/hip_bf16.h>

// ComplexUpSampling2D: 2x2 nearest-neighbor upsample of real+imag f32 tensors.
// Input : x_real, x_imag each [B=16, H=128, W=128, C=64] float32 (channels-last)
// Output: [2, B, 2H, 2W, C] float32 (part 0 = real, part 1 = imag)
//
// Pure data movement (0 FLOPs). HBM floor on MI455X: 134 MB read + 537 MB
// write ~= 671 MB / 23.3 TB/s ~= 29 us. Strategy: load each input float4
// exactly once (linear stream), store it 4x to the 2x2 output block, all with
// 128-bit non-temporal vmem ops so neither the 537 MB output stream nor the
// touch-once input stream pollutes L0/L2.

typedef float v4f __attribute__((ext_vector_type(4)));

namespace {
constexpr unsigned Bn = 16;
constexpr unsigned Hn = 128;
constexpr unsigned Wn = 128;
constexpr unsigned Cn = 64;
constexpr unsigned Ho = Hn * 2;
constexpr unsigned Wo = Wn * 2;
constexpr unsigned CHUNKS = Cn / 4;                 // 16 float4 per pixel
constexpr unsigned PART_THREADS = Bn * Hn * Wn * CHUNKS;  // 4,194,304
constexpr unsigned PART_OUT_Q = Bn * Ho * Wo * CHUNKS;    // float4s per output part
} // namespace

__global__ __launch_bounds__(256) void
complex_upsample2x2_kernel(const float* __restrict__ xr,
                           const float* __restrict__ xi,
                           float* __restrict__ out) {
  const unsigned tid  = blockIdx.x * blockDim.x + threadIdx.x; // chunk id in part
  const unsigned part = blockIdx.y;                            // 0=real, 1=imag

  const float* __restrict__ src = part ? xi : xr;

  // Input is linear in tid: pixel = tid>>4, chunk = tid&15, and
  // pixel*16 + chunk == tid  (since C/4 == 16). One streaming b128 load.
  const v4f v = __builtin_nontemporal_load(
      reinterpret_cast<const v4f*>(src) + tid);

  // Decompose tid -> (b, h, w, chunk) with shifts (all dims are powers of 2).
  const unsigned chunk = tid & 15u;
  const unsigned w     = (tid >> 4)  & 127u;
  const unsigned h     = (tid >> 11) & 127u;
  const unsigned b     =  tid >> 18;

  // Output offset (in float4 units) of pixel (b, 2h, 2w), this chunk.
  const unsigned q00 = (((b * Ho + 2u * h) * Wo + 2u * w) * CHUNKS) + chunk;

  v4f* __restrict__ o = reinterpret_cast<v4f*>(out) + (size_t)part * PART_OUT_Q;

  // 2x2 replication: +CHUNKS = next output column, +Wo*CHUNKS = next row.
  __builtin_nontemporal_store(v, o + q00);
  __builtin_nontemporal_store(v, o + q00 + CHUNKS);
  __builtin_nontemporal_store(v, o + q00 + Wo * CHUNKS);
  __builtin_nontemporal_store(v, o + q00 + Wo * CHUNKS + CHUNKS);
}

extern "C" void kernel_launch(void* const* d_in, const int* in_sizes, int n_in,
                              void* d_out, int out_size, void* d_ws, size_t ws_size,
                              hipStream_t stream) {
  (void)in_sizes; (void)n_in; (void)out_size; (void)d_ws; (void)ws_size;

  const float* xr = reinterpret_cast<const float*>(d_in[0]);
  const float* xi = reinterpret_cast<const float*>(d_in[1]);
  float* out      = reinterpret_cast<float*>(d_out);

  const dim3 block(256, 1, 1);                       // 8 wave32s per block
  const dim3 grid(PART_THREADS / 256u, 2u, 1u);      // y: real / imag part

  complex_upsample2x2_kernel<<<grid, block, 0, stream>>>(xr, xi, out);
}